// SphericalGaborLayer_34376918237395
// MI455X (gfx1250) — compile-verified
//
#include <hip/hip_runtime.h>

typedef float v2f __attribute__((ext_vector_type(2)));
typedef float v8f __attribute__((ext_vector_type(8)));

#define NPTS   65536
#define WID    512
#define LOG2E  1.4426950408889634f
// cos(OMEGA * t) = v_cos(t * OMEGA / (2*pi)),  OMEGA = 0.01
#define CFREQ  0.0015915494309189534f
// exp(-SIGMA^2 * g) = v_exp2(g * -SIGMA^2 * log2(e)),  SIGMA^2 = 0.01
#define CGAUSS (-0.014426950408889634f)

// ---------------------------------------------------------------------------
// Pre-pass: per-column constants (512 columns -> 8 floats each in workspace)
//   [0..2] = r0 row of R = Rz(g)Rx(b)Rz(a)
//   [3..5] = r2 row
//   [6]    = 2*exp(dilate)
//   [7]    = 4*exp(dilate)^2
// ---------------------------------------------------------------------------
__global__ void sgabor_precol(const float* __restrict__ dilate,
                              const float* __restrict__ u,
                              const float* __restrict__ v,
                              const float* __restrict__ w,
                              float* __restrict__ cws) {
    int i = blockIdx.x * blockDim.x + threadIdx.x;
    if (i >= WID) return;
    // alpha = 2*pi*u -> hardware sin/cos take revolutions, so pass u directly
    float ca = __builtin_amdgcn_cosf(u[i]);
    float sa = __builtin_amdgcn_sinf(u[i]);
    float cg = __builtin_amdgcn_cosf(w[i]);
    float sg = __builtin_amdgcn_sinf(w[i]);
    float cb = 2.0f * v[i] - 1.0f;
    cb = fminf(fmaxf(cb, -1.0f + 1e-6f), 1.0f - 1e-6f);   // cos(arccos(clip))
    float sb = __builtin_amdgcn_sqrtf(fmaxf(1.0f - cb * cb, 0.0f)); // beta in [0,pi]
    float* o = cws + i * 8;
    o[0] = cg * ca - sg * cb * sa;
    o[1] = -cg * sa - sg * cb * ca;
    o[2] = sg * sb;
    o[3] = sb * sa;
    o[4] = sb * ca;
    o[5] = cb;
    float twod = 2.0f * __builtin_amdgcn_exp2f(dilate[i] * LOG2E);
    o[6] = twod;
    o[7] = twod * twod;
}

// ---------------------------------------------------------------------------
// Main kernel: one 16x16 output tile per wave via V_WMMA_F32_16X16X4_F32.
//   A (16x4 f32): lanes 0-15 hold M=lane, K=0,1 ; lanes 16-31 hold K=2,3
//   B (4x16 f32): lanes 0-15 hold N=lane, K=0,1 ; lanes 16-31 hold K=2,3
//   C/D (16x16) : vgpr r, lanes 0-15 -> (M=r, N=lane); lanes 16-31 -> (M=r+8)
// ---------------------------------------------------------------------------
__global__ __launch_bounds__(256)
void sgabor_main(const float* __restrict__ inp,   // [N,5] = tp0,tp1,p0,p1,p2
                 const float* __restrict__ Wtp,   // [512,2]
                 const float* __restrict__ btp,   // [512]
                 const float* __restrict__ cws,   // [512,8]
                 float* __restrict__ out) {       // [N,512]
    const int lane = threadIdx.x & 31;
    const int wave = threadIdx.x >> 5;
    const int tile = blockIdx.x * 8 + wave;       // 131072 tiles
    const int tw   = tile & 31;                   // 32 column tiles
    const int tn   = tile >> 5;                   // 4096 row tiles
    const int half = lane >> 4;
    const int m    = lane & 15;
    const int nrow = tn * 16 + m;                 // point row for A
    const int col  = tw * 16 + m;                 // output column for B / elemwise

    // ---- A matrices (point coords + tp), K padded to 4 with zeros ----
    const float* irow = inp + nrow * 5;
    v2f Apts, Atp;
    if (half == 0) {
        Apts.x = irow[2]; Apts.y = irow[3];       // K=0,1 : p0,p1
        Atp.x  = irow[0]; Atp.y  = irow[1];       // K=0,1 : tp0,tp1
    } else {
        Apts.x = irow[4]; Apts.y = 0.0f;          // K=2,3 : p2,0
        Atp.x  = 0.0f;    Atp.y  = 0.0f;          // K=2,3 : 0,0
    }

    // ---- B matrices (r0, r2, W_tp columns) ----
    const float* c = cws + col * 8;
    v2f B0, B2, Bt;
    if (half == 0) {
        B0.x = c[0]; B0.y = c[1];
        B2.x = c[3]; B2.y = c[4];
        Bt.x = Wtp[col * 2 + 0]; Bt.y = Wtp[col * 2 + 1];
    } else {
        B0.x = c[2]; B0.y = 0.0f;
        B2.x = c[5]; B2.y = 0.0f;
        Bt.x = 0.0f; Bt.y = 0.0f;
    }

    v8f zero = {};
    v8f X = __builtin_amdgcn_wmma_f32_16x16x4_f32(false, Apts, false, B0,
                                                  (short)0, zero, false, false);
    v8f Z = __builtin_amdgcn_wmma_f32_16x16x4_f32(false, Apts, false, B2,
                                                  (short)0, zero, false, false);
    float bias = btp[col];
    v8f C = {bias, bias, bias, bias, bias, bias, bias, bias};
    v8f L = __builtin_amdgcn_wmma_f32_16x16x4_f32(false, Atp, false, Bt,
                                                  (short)0, C, false, false);

    const float twod   = c[6];
    const float fourd2 = c[7];
    const int   obase  = (tn * 16 + half * 8) * WID + col;

#pragma unroll
    for (int r = 0; r < 8; ++r) {
        float x   = X[r];
        float z   = Z[r];
        float lin = L[r];
        float inv   = __builtin_amdgcn_rcpf(1.000001f + z);   // 1/(eps+1+z)
        float freq  = twod * x * inv + lin;
        float gauss = fourd2 * (1.0f - z) * inv + lin * lin;
        float val = __builtin_amdgcn_cosf(freq * CFREQ)
                  * __builtin_amdgcn_exp2f(gauss * CGAUSS);
        out[obase + r * WID] = val;
    }
}

extern "C" void kernel_launch(void* const* d_in, const int* in_sizes, int n_in,
                              void* d_out, int out_size, void* d_ws, size_t ws_size,
                              hipStream_t stream) {
    const float* inp = (const float*)d_in[0];   // [N,5]
    const float* dil = (const float*)d_in[1];   // [1,512]
    const float* u   = (const float*)d_in[2];   // [512]
    const float* v   = (const float*)d_in[3];   // [512]
    const float* w   = (const float*)d_in[4];   // [512]
    const float* Wtp = (const float*)d_in[5];   // [512,2]
    const float* btp = (const float*)d_in[6];   // [512]
    float* cws = (float*)d_ws;                  // 512*8 floats = 16 KB
    float* out = (float*)d_out;

    sgabor_precol<<<2, 256, 0, stream>>>(dil, u, v, w, cws);
    // 131072 tiles (4096 n-tiles x 32 w-tiles), 8 waves per 256-thread block
    sgabor_main<<<16384, 256, 0, stream>>>(inp, Wtp, btp, cws, out);
}